// MultiModalFourier_1047972020965
// MI455X (gfx1250) — compile-verified
//
#include <hip/hip_runtime.h>
#include <hip/hip_bf16.h>
#include <math.h>

// Problem constants from the reference
#define NB 8
#define NS 4096
#define ND 128
#define NH 40
#define NMOD 3

typedef __attribute__((ext_vector_type(2))) float v2f;
typedef __attribute__((ext_vector_type(8))) float v8f;

// ---------------------------------------------------------------------------
// Kernel 1: LayerNorm. One wave32 per (modality,batch,seq) row of 128 floats.
// Each lane owns a float4 (4 channels). Population variance (ddof=0), eps=1e-5.
// ---------------------------------------------------------------------------
__global__ __launch_bounds__(256) void mmf_layernorm_kernel(
    const float* __restrict__ xt, const float* __restrict__ xa,
    const float* __restrict__ xv, const float* __restrict__ g,
    const float* __restrict__ bn, float* __restrict__ xn) {
  const int lane = threadIdx.x & 31;
  const long row = (long)blockIdx.x * 8 + (threadIdx.x >> 5);  // 0..3*B*S-1
  const long RPM = (long)NB * NS;
  const int m = (int)(row / RPM);
  const long rloc = row - (long)m * RPM;
  const float* xs = (m == 0) ? xt : (m == 1) ? xa : xv;

  float4 x = ((const float4*)xs)[rloc * 32 + lane];
  float s = x.x + x.y + x.z + x.w;
  float q = x.x * x.x + x.y * x.y + x.z * x.z + x.w * x.w;
#pragma unroll
  for (int off = 16; off > 0; off >>= 1) {
    s += __shfl_xor(s, off, 32);
    q += __shfl_xor(q, off, 32);
  }
  const float mean = s * (1.0f / 128.0f);
  const float var = q * (1.0f / 128.0f) - mean * mean;
  const float inv = rsqrtf(var + 1e-5f);

  float4 gg = ((const float4*)g)[lane];
  float4 bb = ((const float4*)bn)[lane];
  float4 y;
  y.x = (x.x - mean) * inv * gg.x + bb.x;
  y.y = (x.y - mean) * inv * gg.y + bb.y;
  y.z = (x.z - mean) * inv * gg.z + bb.z;
  y.w = (x.w - mean) * inv * gg.w + bb.w;
  ((float4*)xn)[row * 32 + lane] = y;
}

// ---------------------------------------------------------------------------
// Kernel 2: column sums of xn over the time axis (full S, as in reference).
// One block per (modality,batch); thread d walks the column. Deterministic.
// xn is L2-resident (50MB << 192MB); prefetch the stream ahead.
// ---------------------------------------------------------------------------
__global__ __launch_bounds__(128) void mmf_colsum_kernel(
    const float* __restrict__ xn, float* __restrict__ csum) {
  const int mb = blockIdx.x;   // m*NB + i, 0..23
  const int d = threadIdx.x;   // 0..127
  const float* p = xn + (long)mb * NS * ND + d;
  float s = 0.0f;
  for (int n = 0; n < NS; ++n) {
    if ((n & 31) == 0 && n + 64 < NS)
      __builtin_prefetch(p + (long)(n + 64) * ND, 0, 1);  // global_prefetch_b8
    s += p[(long)n * ND];
  }
  csum[mb * ND + d] = s;
}

// ---------------------------------------------------------------------------
// Kernel 3: 18 tiny MLPs on V_WMMA_F32_16X16X4_F32. One wave32 per MLP.
// Stage 0 (cooperative, branch-divergence only in the preload):
//   LDS aLds [16][128] : rows 0..7 = csum/len, rows 8..15 = 0
//   LDS w1Lds[128][48] : cols 0..39 = w1, cols 40..47 = 0
//   LDS b1Lds[48]      : 0..39 = b1, 40..47 = 0
//   LDS w2Lds[48][128] : rows 0..39 = w2, rows 40..47 = 0
// Stage 1: GEMM1 (16x128 @ 128x48) -> C-layout accs, entirely LDS-fed, no
//   branches, EXEC all-ones through every WMMA.
// Stage 2: +b1, exact GELU (pad lanes compute gelu(0)=0), write H1 to LDS.
// Stage 3: GEMM2 (16x48 @ 48x128), LDS-fed, branch-free.
// Stage 4: +b2, store valid rows 0..7.
// Fragment layouts (ISA 7.12.2, 32-bit, wave32):
//   A 16x4 : lane L -> row L&15 ; VGPR0/1 = K = (L>>4)*2 + {0,1}
//   B 4x16 : lanes 0-15 cols, VGPR0/1 = K {0,1}; lanes 16-31 -> K {2,3}
//   C/D    : VGPR r -> (M=r, N=lane) lanes 0-15 ; (M=r+8) lanes 16-31
// ---------------------------------------------------------------------------
struct MlpArgs {
  const float* b1[18];
  const float* b2[18];
  const float* w1[18];
  const float* w2[18];
  const int* lens[3];
  int csrc[18];
};

__device__ inline v8f zero8() {
  v8f z;
#pragma unroll
  for (int i = 0; i < 8; ++i) z[i] = 0.0f;
  return z;
}

__global__ __launch_bounds__(32) void mmf_wmma_mlp_kernel(
    MlpArgs A, const float* __restrict__ csum, float* __restrict__ mlp_out) {
  const int job = blockIdx.x;
  const int lane = threadIdx.x;
  const int row = lane & 15;            // M row / N col within a 16-tile
  const int koff = (lane >> 4) * 2;     // K sub-offset for A/B fragments

  const int src = A.csrc[job];
  const float* c = csum + src * (NB * ND);
  const int* ln = A.lens[src];
  const float* w1 = A.w1[job];
  const float* b1 = A.b1[job];
  const float* w2 = A.w2[job];
  const float* b2 = A.b2[job];

  __shared__ float aLds[16 * 128];
  __shared__ float w1Lds[128 * 48];
  __shared__ float b1Lds[48];
  __shared__ float w2Lds[48 * 128];
  __shared__ float h1[16 * 48];

  // ---- Stage 0: zero-fill pads, then cooperative loads ----
  for (int i = lane; i < 16 * 128; i += 32) aLds[i] = 0.0f;
  for (int i = lane; i < 128 * 48; i += 32) w1Lds[i] = 0.0f;
  for (int i = lane; i < 48 * 128; i += 32) w2Lds[i] = 0.0f;
  if (lane < 48) b1Lds[lane < 48 ? lane : 0] = 0.0f;  // lane 0..31 covers 0..31
  for (int i = lane + 32; i < 48; i += 32) b1Lds[i] = 0.0f;

  for (int i = lane; i < NB * ND; i += 32) {          // A rows 0..7, scaled
    const int r = i >> 7;
    aLds[i] = c[i] * (1.0f / (float)ln[r]);
  }
  for (int i = lane; i < ND * NH; i += 32) {          // w1 -> padded stride 48
    const int k = i / NH;
    const int col = i - k * NH;
    w1Lds[k * 48 + col] = w1[i];
  }
  for (int i = lane; i < NH; i += 32) b1Lds[i] = b1[i];
  for (int i = lane; i < NH * ND; i += 32) w2Lds[i] = w2[i];  // rows 0..39
  __syncthreads();

  // ---- Stage 1: GEMM1 (branch-free, LDS-fed) ----
  v8f acc[3];
#pragma unroll
  for (int t = 0; t < 3; ++t) acc[t] = zero8();

  for (int kc = 0; kc < ND / 4; ++kc) {
    const int k0 = kc * 4 + koff;
    v2f af;
    af.x = aLds[row * 128 + k0];
    af.y = aLds[row * 128 + k0 + 1];
#pragma unroll
    for (int t = 0; t < 3; ++t) {
      v2f bf;
      bf.x = w1Lds[k0 * 48 + t * 16 + row];
      bf.y = w1Lds[(k0 + 1) * 48 + t * 16 + row];
      acc[t] = __builtin_amdgcn_wmma_f32_16x16x4_f32(
          false, af, false, bf, (short)0, acc[t], false, false);
    }
  }

  // ---- Stage 2: +b1, exact GELU, C-layout -> plain [16][48] in LDS ----
  const int mbase = (lane >> 4) * 8;
#pragma unroll
  for (int t = 0; t < 3; ++t) {
    const int col = t * 16 + row;
    const float bb = b1Lds[col];
#pragma unroll
    for (int r = 0; r < 8; ++r) {
      const float x = acc[t][r] + bb;                 // pad cols: 0 + 0
      const float gelu = 0.5f * x * (1.0f + erff(x * 0.70710678118654752f));
      h1[(mbase + r) * 48 + col] = gelu;              // gelu(0)=0 in pads
    }
  }
  __syncthreads();

  // ---- Stage 3: GEMM2 (branch-free, LDS-fed) ----
  v8f acc2[8];
#pragma unroll
  for (int t = 0; t < 8; ++t) acc2[t] = zero8();

  for (int kc = 0; kc < 48 / 4; ++kc) {
    const int k0 = kc * 4 + koff;
    v2f af;
    af.x = h1[row * 48 + k0];
    af.y = h1[row * 48 + k0 + 1];
#pragma unroll
    for (int t = 0; t < 8; ++t) {
      v2f bf;
      bf.x = w2Lds[k0 * 128 + t * 16 + row];
      bf.y = w2Lds[(k0 + 1) * 128 + t * 16 + row];
      acc2[t] = __builtin_amdgcn_wmma_f32_16x16x4_f32(
          false, af, false, bf, (short)0, acc2[t], false, false);
    }
  }

  // ---- Stage 4: +b2, store rows 0..7 (lanes 0-15 hold M=0..7, VGPRs 0..7) ----
  float* out = mlp_out + job * (NB * ND);
  if (lane < 16) {
#pragma unroll
    for (int t = 0; t < 8; ++t) {
      const int col = t * 16 + lane;
      const float bb = b2[col];
#pragma unroll
      for (int r = 0; r < 8; ++r) out[r * ND + col] = acc2[t][r] + bb;
    }
  }
}

// ---------------------------------------------------------------------------
// Kernel 4: combine the 18 MLP outputs into per-modality scale S and offset O.
//   S_m = 2 + 0.5*(W1+W2) + Ww ;  O_m = 0.5*(B1+B2) + Wb
// ---------------------------------------------------------------------------
__global__ __launch_bounds__(256) void mmf_combine_kernel(
    const float* __restrict__ mlp_out, float* __restrict__ Sc,
    float* __restrict__ Oc) {
  const int i = blockIdx.x * blockDim.x + threadIdx.x;
  if (i >= NMOD * NB * ND) return;
  const int m = i / (NB * ND);
  const int p = i % (NB * ND);
  int w1i, w2i, b1i, b2i, swi, sbi;
  if (m == 0)      { w1i = 0; w2i = 1; b1i = 2;  b2i = 3;  swi = 12; sbi = 13; }
  else if (m == 1) { w1i = 4; w2i = 5; b1i = 6;  b2i = 7;  swi = 14; sbi = 15; }
  else             { w1i = 8; w2i = 9; b1i = 10; b2i = 11; swi = 16; sbi = 17; }
  const int stride = NB * ND;
  Sc[i] = 2.0f + 0.5f * (mlp_out[w1i * stride + p] + mlp_out[w2i * stride + p])
               + mlp_out[swi * stride + p];
  Oc[i] = 0.5f * (mlp_out[b1i * stride + p] + mlp_out[b2i * stride + p])
               + mlp_out[sbi * stride + p];
}

// ---------------------------------------------------------------------------
// Kernel 5: finalize (the collapsed FFT filter).
//   out = x + [n<l]*( S[i,d]*xn + [n==0]*O[i,d] )
// float4 per thread; fully coalesced; xn/x hot in L2.
// ---------------------------------------------------------------------------
__global__ __launch_bounds__(256) void mmf_finalize_kernel(
    const float* __restrict__ xt, const float* __restrict__ xa,
    const float* __restrict__ xv, const int* __restrict__ lt,
    const int* __restrict__ la, const int* __restrict__ lv,
    const float* __restrict__ xn, const float* __restrict__ Sc,
    const float* __restrict__ Oc, float* __restrict__ out) {
  const long QPM = (long)NB * NS * (ND / 4);  // float4s per modality
  const long idx = (long)blockIdx.x * blockDim.x + threadIdx.x;
  if (idx >= NMOD * QPM) return;
  const int m = (int)(idx / QPM);
  const long r = idx - (long)m * QPM;
  const int i = (int)(r / ((long)NS * (ND / 4)));
  const long rr = r - (long)i * NS * (ND / 4);
  const int n = (int)(rr >> 5);   // / (ND/4)
  const int dq = (int)(rr & 31);

  const float* xs = (m == 0) ? xt : (m == 1) ? xa : xv;
  const int* ls = (m == 0) ? lt : (m == 1) ? la : lv;

  float4 x = ((const float4*)xs)[r];
  float4 res = x;
  const int l = ls[i];
  if (n < l) {
    float4 z = ((const float4*)xn)[idx];
    const int base = (m * NB + i) * ND + dq * 4;
    float4 s = *(const float4*)(Sc + base);
    res.x += s.x * z.x;
    res.y += s.y * z.y;
    res.z += s.z * z.z;
    res.w += s.w * z.w;
    if (n == 0) {
      float4 o = *(const float4*)(Oc + base);
      res.x += o.x; res.y += o.y; res.z += o.z; res.w += o.w;
    }
  }
  ((float4*)out)[idx] = res;
}

// ---------------------------------------------------------------------------
// Host-side launch. Input index map assumes jax pytree (sorted-key) expansion
// of the nested params dict after the 6 positional arrays:
//   0:xt 1:xa 2:xv 3:len_t 4:len_a 5:len_v
//   a2a.Wb:6-9  a2a.Ww:10-13  norm_b:14  norm_g:15
//   o2a.B1:16 B2:20 W1:24 W2:28 | o2t.B1:32 B2:36 W1:40 W2:44
//   o2v.B1:48 B2:52 W1:56 W2:60 | t2t.Wb:64 Ww:68 | v2v.Wb:72 Ww:76
//   (each MLP leaf group sorted: b1,b2,w1,w2)
// ---------------------------------------------------------------------------
extern "C" void kernel_launch(void* const* d_in, const int* in_sizes, int n_in,
                              void* d_out, int out_size, void* d_ws,
                              size_t ws_size, hipStream_t stream) {
  (void)in_sizes; (void)n_in; (void)out_size; (void)ws_size;

  const float* xt = (const float*)d_in[0];
  const float* xa = (const float*)d_in[1];
  const float* xv = (const float*)d_in[2];
  const int* lt = (const int*)d_in[3];
  const int* la = (const int*)d_in[4];
  const int* lv = (const int*)d_in[5];
  const float* norm_b = (const float*)d_in[14];
  const float* norm_g = (const float*)d_in[15];

  // MLP job table: {param-group base input index, c-source modality}
  // jobs: o2t{W1,W2,B1,B2} o2a{...} o2v{...} t2t{Ww,Wb} a2a{Ww,Wb} v2v{Ww,Wb}
  static const int kBase[18] = {40, 44, 32, 36,  24, 28, 16, 20,
                                56, 60, 48, 52,  68, 64, 10, 6, 76, 72};
  static const int kCsrc[18] = {1, 2, 1, 2,  0, 2, 0, 2,
                                0, 1, 0, 1,  0, 0, 1, 1, 2, 2};

  // Workspace layout (floats)
  float* ws = (float*)d_ws;
  float* xn = ws;                                     // 3*B*S*D
  float* csum = xn + (long)NMOD * NB * NS * ND;       // 3*B*D
  float* mlp_out = csum + NMOD * NB * ND;             // 18*B*D
  float* Sc = mlp_out + 18 * NB * ND;                 // 3*B*D
  float* Oc = Sc + NMOD * NB * ND;                    // 3*B*D

  // 1) LayerNorm all three modalities (one wave per row, 8 rows per block)
  {
    const long rows = (long)NMOD * NB * NS;           // 98304
    mmf_layernorm_kernel<<<(int)(rows / 8), 256, 0, stream>>>(
        xt, xa, xv, norm_g, norm_b, xn);
  }

  // 2) Column sums (means numerators)
  mmf_colsum_kernel<<<NMOD * NB, 128, 0, stream>>>(xn, csum);

  // 3) 18 WMMA MLPs
  {
    MlpArgs margs;
    for (int j = 0; j < 18; ++j) {
      margs.b1[j] = (const float*)d_in[kBase[j] + 0];
      margs.b2[j] = (const float*)d_in[kBase[j] + 1];
      margs.w1[j] = (const float*)d_in[kBase[j] + 2];
      margs.w2[j] = (const float*)d_in[kBase[j] + 3];
      margs.csrc[j] = kCsrc[j];
    }
    margs.lens[0] = lt; margs.lens[1] = la; margs.lens[2] = lv;
    mmf_wmma_mlp_kernel<<<18, 32, 0, stream>>>(margs, csum, mlp_out);
  }

  // 4) Combine into per-modality scale/offset
  {
    const int tot = NMOD * NB * ND;
    mmf_combine_kernel<<<(tot + 255) / 256, 256, 0, stream>>>(mlp_out, Sc, Oc);
  }

  // 5) Finalize (collapsed spectral filter + residual)
  {
    const long quads = (long)NMOD * NB * NS * (ND / 4);  // 3145728
    mmf_finalize_kernel<<<(int)((quads + 255) / 256), 256, 0, stream>>>(
        xt, xa, xv, lt, la, lv, xn, Sc, Oc, (float*)d_out);
  }
}